// PQLinear_15350213116641
// MI455X (gfx1250) — compile-verified
//
#include <hip/hip_runtime.h>
#include <hip/hip_bf16.h>

typedef __attribute__((ext_vector_type(2))) float v2f;
typedef __attribute__((ext_vector_type(8))) float v8f;

#if __has_builtin(__builtin_amdgcn_global_load_async_to_lds_b128) && \
    __has_builtin(__builtin_amdgcn_s_wait_asynccnt)
#define PQ_USE_ASYNC_LDS 1
typedef int v4i __attribute__((ext_vector_type(4)));
typedef __attribute__((address_space(1))) v4i g_v4i;  // global (AS1) int4
typedef __attribute__((address_space(3))) v4i l_v4i;  // LDS (AS3) int4
#endif

namespace {
constexpr int C   = 32;       // codebooks
constexpr int K   = 16;       // centroids per codebook
constexpr int D   = 128;      // subvector dim
constexpr int INF = C * D;    // 4096 input features
constexpr int O   = 2048;     // output features
constexpr int N   = 2 * 1024; // tokens (BATCH*SEQ)
constexpr int OT  = 64;       // o-tile held in LDS by gather kernel
constexpr int TT  = 64;       // tokens per gather block
}

// ---------------------------------------------------------------------------
// Kernel 1: lut[c][k][o] = sum_d centroids[c][k][d] * weight[c][d][o]
// One wave per (codebook, 16-wide o tile); chain of 32 V_WMMA_F32_16X16X4_F32.
// A fragment (16x4 f32): lane row = lane&15, halves hold d-pairs {0,1}/{2,3}.
// ---------------------------------------------------------------------------
__global__ __launch_bounds__(32) void pq_build_lut(
    const float* __restrict__ centroids, const float* __restrict__ weight,
    float* __restrict__ lut) {
  const int ot   = blockIdx.x * 16;
  const int c    = blockIdx.y;
  const int lane = threadIdx.x;
  const int row  = lane & 15;
  const int hi   = lane >> 4;
  const float* A = centroids + (size_t)c * K * D;  // [16][128]
  const float* B = weight    + (size_t)c * D * O;  // [128][2048]
  v8f acc = {};
  for (int d0 = 0; d0 < D; d0 += 4) {
    const int dd = d0 + 2 * hi;
    v2f a, b;
    a.x = A[row * D + dd];
    a.y = A[row * D + dd + 1];
    b.x = B[(size_t)dd * O + ot + row];
    b.y = B[(size_t)(dd + 1) * O + ot + row];
    acc = __builtin_amdgcn_wmma_f32_16x16x4_f32(false, a, false, b, (short)0,
                                                acc, false, false);
  }
  // D layout: VGPR v -> M = v + 8*hi, N = row
  float* dst = lut + (size_t)c * K * O + ot + row;
#pragma unroll
  for (int v = 0; v < 8; ++v) dst[(size_t)(v + 8 * hi) * O] = acc[v];
}

// ---------------------------------------------------------------------------
// Kernel 2: idx[c][n] = argmin_k ( c2[c][k] - 2 * x[n,c,:].centroids[c][k,:] )
// (x^2 term is k-invariant; sqrt is monotone -> both dropped.)
// One wave per (codebook, 16-token tile): 32 f32 WMMAs + shuffle argmin.
// ---------------------------------------------------------------------------
__global__ __launch_bounds__(256) void pq_assign(
    const float* __restrict__ x, const float* __restrict__ centroids,
    int* __restrict__ idx) {
  const int lane = threadIdx.x & 31;
  const int wave = threadIdx.x >> 5;
  const int c    = blockIdx.y * 8 + wave;
  const int n0   = blockIdx.x * 16;
  const int row  = lane & 15;
  const int hi   = lane >> 4;
  const float* X  = x + (size_t)n0 * INF + (size_t)c * D;  // token-major rows
  const float* Cc = centroids + (size_t)c * K * D;         // [16][128]
  v8f acc = {};
  float c2p = 0.f;  // partial ||centroid_row||^2 over this lane's d-slices
  for (int d0 = 0; d0 < D; d0 += 4) {
    const int dd = d0 + 2 * hi;
    v2f a, b;
    a.x = X[(size_t)row * INF + dd];       // A: tokens x d
    a.y = X[(size_t)row * INF + dd + 1];
    b.x = Cc[row * D + dd];                // B: d x centroids (k = row)
    b.y = Cc[row * D + dd + 1];
    c2p += b.x * b.x + b.y * b.y;
    acc = __builtin_amdgcn_wmma_f32_16x16x4_f32(false, a, false, b, (short)0,
                                                acc, false, false);
  }
  const float c2 = c2p + __shfl_xor(c2p, 16, 32);  // full sum_d for k = row
#pragma unroll
  for (int v = 0; v < 8; ++v) {
    // D fragment: token m = v + 8*hi, k = row (= lane%16)
    float s  = c2 - 2.f * acc[v];
    int   bk = row;
#pragma unroll
    for (int m = 1; m <= 8; m <<= 1) {  // butterfly within 16-lane half
      const float so = __shfl_xor(s, m, 32);
      const int   ko = __shfl_xor(bk, m, 32);
      if (so < s || (so == s && ko < bk)) { s = so; bk = ko; }
    }
    if (row == 0) idx[(size_t)c * N + n0 + v + 8 * hi] = bk;
  }
}

// ---------------------------------------------------------------------------
// Kernel 3: out[n][o] = bias[o] + sum_c lut[c][idx[c][n]][o]
// LDS-tiled gather: 128 KB LUT slice (all c,k for a 64-wide o tile) is
// loaded once — via async global->LDS DMA (ASYNCcnt path) when available —
// and reused by 64 tokens, so the dominant N*O*C gather volume comes from
// LDS instead of L2/HBM.
// ---------------------------------------------------------------------------
__global__ __launch_bounds__(256) void pq_gather(
    const float* __restrict__ lut, const int* __restrict__ idx,
    const float* __restrict__ bias, float* __restrict__ out) {
  __shared__ float slut[C * K * OT];  // 32*16*64*4 = 128 KB
  __shared__ int   sidx[TT * C];      // 8 KB
  const int o0  = blockIdx.x * OT;
  const int n0  = blockIdx.y * TT;
  const int tid = threadIdx.x;

#if defined(PQ_USE_ASYNC_LDS)
  // Async fill: 16 B per lane per op; 256 threads x 32 ops = 128 KB.
  // Bypasses VGPRs entirely; tracked by ASYNCcnt.
#pragma unroll
  for (int it = 0; it < (C * K * OT / 4) / 256; ++it) {
    const int f4  = tid + it * 256;        // float4 slot
    const int oo4 = f4 & (OT / 4 - 1);     // float4 col within o tile
    const int rr  = f4 >> 4;               // row = cc*K + kk (0..511)
    const float* gsrc = lut + (size_t)rr * O + o0 + oo4 * 4;
    __builtin_amdgcn_global_load_async_to_lds_b128(
        (g_v4i*)gsrc, (l_v4i*)(slut + (size_t)f4 * 4), 0, 0);
  }
#else
  for (int f = tid; f < C * K * OT; f += 256) {
    const int cc = f / (K * OT);
    const int kk = (f / OT) & (K - 1);
    const int oo = f & (OT - 1);
    slut[f] = lut[((size_t)cc * K + kk) * O + o0 + oo];
  }
#endif
  for (int f = tid; f < TT * C; f += 256) {
    const int t  = f >> 5;
    const int cc = f & 31;
    sidx[f] = idx[(size_t)cc * N + n0 + t];
  }
#if defined(PQ_USE_ASYNC_LDS)
  __builtin_amdgcn_s_wait_asynccnt(0);
#endif
  __syncthreads();

  const int col  = tid & 15;   // float4 column within o tile (OT/4 = 16)
  const int tsub = tid >> 4;   // 16 tokens per pass
  const float4 b4 = ((const float4*)(bias + o0))[col];
  for (int tt = 0; tt < TT; tt += 16) {
    const int t = tt + tsub;
    float4 acc = b4;
#pragma unroll
    for (int cc = 0; cc < C; ++cc) {
      const int    kk = sidx[t * C + cc];
      const float4 lv = ((const float4*)(slut + (cc * K + kk) * OT))[col];
      acc.x += lv.x; acc.y += lv.y; acc.z += lv.z; acc.w += lv.w;
    }
    ((float4*)(out + (size_t)(n0 + t) * O + o0))[col] = acc;
  }
}

extern "C" void kernel_launch(void* const* d_in, const int* in_sizes, int n_in,
                              void* d_out, int out_size, void* d_ws, size_t ws_size,
                              hipStream_t stream) {
  (void)in_sizes; (void)n_in; (void)out_size; (void)ws_size;
  const float* x         = (const float*)d_in[0];
  const float* centroids = (const float*)d_in[1];
  const float* weight    = (const float*)d_in[2];
  // d_in[3] (inverse_temperature_logit) only affects gradients, not fwd value
  const float* bias      = (const float*)d_in[4];
  float* out = (float*)d_out;

  // workspace: lut (4 MB) | idx (256 KB)
  float* lut = (float*)d_ws;
  int*   idx = (int*)((char*)d_ws + (size_t)C * K * O * sizeof(float));

  pq_build_lut<<<dim3(O / 16, C), 32, 0, stream>>>(centroids, weight, lut);
  pq_assign<<<dim3(N / 16, C / 8), 256, 0, stream>>>(x, centroids, idx);
  pq_gather<<<dim3(O / OT, N / TT), 256, 0, stream>>>(lut, idx, bias, out);
}